// CRFLayer_34720515621428
// MI455X (gfx1250) — compile-verified
//
#include <hip/hip_runtime.h>
#include <stdint.h>

namespace {
constexpr int kB = 128;   // batch
constexpr int kS = 512;   // sequence length
constexpr int kT = 256;   // tags
constexpr int kPad = 4;                  // padding floats per transposed row
constexpr int kRow = kT + kPad;          // 260 floats per LDS row (1040B, 16B aligned)
constexpr int kHalves = 2;               // i-dimension split across thread halves
constexpr int kBlk = kT * kHalves;       // 512 threads = 16 waves (wave32)
constexpr int kIChunk = kT / kHalves;    // 128 i's per thread
}

// gfx1250 async global->LDS copy (ASYNCcnt path). The low 32 bits of a
// generic pointer into LDS are the LDS byte offset (aperture mapping).
__device__ __forceinline__ void async_copy_g2l_b32(const void* g, void* l) {
  uint32_t lds_off = (uint32_t)(uintptr_t)l;
  uint64_t gaddr   = (uint64_t)(uintptr_t)g;
  asm volatile("global_load_async_to_lds_b32 %0, %1, off"
               :: "v"(lds_off), "v"(gaddr)
               : "memory");
}

__device__ __forceinline__ void wait_async_zero() {
  asm volatile("s_wait_asynccnt 0" ::: "memory");
}

__global__ __launch_bounds__(kBlk, 1) void viterbi_forward(
    const float* __restrict__ em,       // [B,S,T]
    const uint8_t* __restrict__ mask,   // [B,S] (bool bytes)
    const float* __restrict__ trans,    // [T,T]
    const float* __restrict__ startt,   // [T]
    const float* __restrict__ endt,     // [T]
    int* __restrict__ paths,            // [B,S] (only [S-1] written here)
    float* __restrict__ scores,         // [B]
    uint8_t* __restrict__ bp)           // [B,S,T] backpointers
{
  extern __shared__ float tlT[];        // kT * kRow floats: trans transposed+padded
  __shared__ float sbuf[2][kT];         // double-buffered score vector
  __shared__ float embuf[2][kT];        // double-buffered emissions (async-filled)
  __shared__ float pbv[kHalves][kT];    // partial best values
  __shared__ int   pbi[kHalves][kT];    // partial best indices
  __shared__ float red_v[kT];
  __shared__ int   red_i[kT];

  const int b     = blockIdx.x;
  const int tid   = threadIdx.x;
  const int j     = tid & (kT - 1);     // output tag
  const int half  = tid >> 8;           // which i-half this thread reduces
  const int ibase = half * kIChunk;     // 0 or 128

  // Async-copy transitions (row-major in global) into transposed padded LDS:
  // tlT[j*kRow + i] = trans[i*kT + j]. Coalesced global reads per row.
  for (int i = ibase; i < ibase + kIChunk; ++i) {
    async_copy_g2l_b32(&trans[i * kT + j], &tlT[j * kRow + i]);
  }
  wait_async_zero();

  // score_0 = start + emissions[:,0]; prefetch emissions for t=1 async.
  float my_score = 0.0f;
  if (half == 0) {
    my_score = startt[j] + em[((size_t)b * kS + 0) * kT + j];
    async_copy_g2l_b32(&em[((size_t)b * kS + 1) * kT + j], &embuf[1][j]);
  }
  __syncthreads();  // every thread waited its own asyncs -> tlT fully valid

  int buf = 0;
  for (int t = 1; t < kS; ++t) {
    if (half == 0) sbuf[buf][j] = my_score;
    wait_async_zero();               // embuf[t&1][j] has landed (half 0's async)
    __syncthreads();                 // sbuf[buf] visible to all threads

    if (half == 0 && t + 1 < kS) {
      async_copy_g2l_b32(&em[((size_t)b * kS + t + 1) * kT + j],
                         &embuf[(t + 1) & 1][j]);
    }

    // Max-plus partial reduction over this thread's i-half.
    // ds_load_b128 for trans column chunk (padded -> minimal conflicts),
    // broadcast ds_load_b128 for the shared score vector; float4 add
    // encourages v_pk_add_f32 (VOP3P packed f32).
    const float4* tcol  = (const float4*)&tlT[j * kRow + ibase];
    const float4* srow4 = (const float4*)&sbuf[buf][ibase];
    float best = -__builtin_inff();
    int bi = ibase;
#pragma unroll 8
    for (int q = 0; q < kIChunk / 4; ++q) {
      float4 tv = tcol[q];
      float4 sv = srow4[q];
      float4 vs = tv + sv;   // packed adds
      const int i0 = ibase + 4 * q;
      if (vs.x > best) { best = vs.x; bi = i0 + 0; }
      if (vs.y > best) { best = vs.y; bi = i0 + 1; }
      if (vs.z > best) { best = vs.z; bi = i0 + 2; }
      if (vs.w > best) { best = vs.w; bi = i0 + 3; }
    }
    pbv[half][j] = best;
    pbi[half][j] = bi;
    __syncthreads();

    if (half == 0) {
      // Combine halves; strict '>' keeps lower-half index on ties
      // (first-max semantics, matching jnp.argmax).
      float b0 = pbv[0][j]; int i0 = pbi[0][j];
      float b1 = pbv[1][j]; int i1 = pbi[1][j];
      if (b1 > b0) { b0 = b1; i0 = i1; }

      const bool m  = mask[b * kS + t] != 0;
      const float ns = b0 + embuf[t & 1][j];
      my_score = m ? ns : my_score;             // masked step carries score
      const int nbp = m ? i0 : j;               // identity backpointer if masked
      bp[((size_t)b * kS + t) * kT + j] = (uint8_t)nbp;
    }
    buf ^= 1;
  }

  // Add end transitions; block-wide argmax with first-max tie-breaking.
  if (half == 0) {
    red_v[j] = my_score + endt[j];
    red_i[j] = j;
  }
  __syncthreads();
  for (int s = kT / 2; s > 0; s >>= 1) {
    if (tid < s) {
      float vr = red_v[tid + s]; int ir = red_i[tid + s];
      float vl = red_v[tid];     int il = red_i[tid];
      if (vr > vl || (vr == vl && ir < il)) { red_v[tid] = vr; red_i[tid] = ir; }
    }
    __syncthreads();
  }
  if (tid == 0) {
    scores[b] = red_v[0];
    paths[b * kS + (kS - 1)] = red_i[0];
  }
}

__global__ void viterbi_backtrack(const uint8_t* __restrict__ bp,
                                  int* __restrict__ paths) {
  const int b = blockIdx.x * blockDim.x + threadIdx.x;
  if (b >= kB) return;
  int tag = paths[b * kS + (kS - 1)];
  for (int t = kS - 1; t >= 1; --t) {
    tag = bp[((size_t)b * kS + t) * kT + tag];
    paths[b * kS + (t - 1)] = tag;
  }
}

extern "C" void kernel_launch(void* const* d_in, const int* in_sizes, int n_in,
                              void* d_out, int out_size, void* d_ws, size_t ws_size,
                              hipStream_t stream) {
  (void)in_sizes; (void)n_in; (void)out_size; (void)ws_size;
  const float*   em     = (const float*)d_in[0];
  const uint8_t* mask   = (const uint8_t*)d_in[1];
  const float*   trans  = (const float*)d_in[2];
  const float*   startt = (const float*)d_in[3];
  const float*   endt   = (const float*)d_in[4];

  int*   paths  = (int*)d_out;                 // [B,S] int32
  float* scores = (float*)d_out + kB * kS;     // [B] f32
  uint8_t* bp   = (uint8_t*)d_ws;              // [B,S,T] u8 = 16 MB

  const size_t dyn_lds = (size_t)kT * kRow * sizeof(float);  // 266,240 B
  viterbi_forward<<<dim3(kB), dim3(kBlk), dyn_lds, stream>>>(
      em, mask, trans, startt, endt, paths, scores, bp);
  viterbi_backtrack<<<dim3(4), dim3(32), 0, stream>>>(bp, paths);
}